// EdgeRNNCell_1752346656989
// MI455X (gfx1250) — compile-verified
//
#include <hip/hip_runtime.h>
#include <math.h>

#define NB   32      // batches
#define NT   8       // timesteps
#define NPTS 1024    // points
#define KNN  20
#define HD   64      // hidden
#define CLS  14
#define EPSF 1e-5f

typedef __attribute__((ext_vector_type(2))) float v2f;
typedef __attribute__((ext_vector_type(8))) float v8f;

// ---------------------------------------------------------------------------
// zero the per-(t,b) channel-max accumulator: 7*32*64 floats
// ---------------------------------------------------------------------------
__global__ __launch_bounds__(256) void init_acc_kernel(float* __restrict__ acc) {
    int i = blockIdx.x * 256 + threadIdx.x;
    if (i < (NT - 1) * NB * HD) acc[i] = 0.0f;
}

// ---------------------------------------------------------------------------
// EdgeConv: distances via V_WMMA_F32_16X16X4_F32, top-K via wave32 shuffles,
// 6->64 edge MLP + BN, channel max into acc[t][b][64]
// grid: (64 row tiles, 32 batches, 7 timesteps), block: 128 (4 waves)
// ---------------------------------------------------------------------------
__global__ __launch_bounds__(128) void edge_kernel(
    const float* __restrict__ x,     // [B,T,N,3]
    const float* __restrict__ W1,    // [6,64]
    const float* __restrict__ b1,    // [64]
    const float* __restrict__ g1,    // [64]
    const float* __restrict__ be1,   // [64]
    float* __restrict__ acc)         // [7][32][64]
{
    __shared__ float dotTile[16 * NPTS];   // 64 KB: dot(center_m, neigh_col)

    const int tileRow = blockIdx.x;        // 0..63 -> rows tileRow*16 ..
    const int b       = blockIdx.y;        // 0..31
    const int t       = blockIdx.z;        // 0..6 ; xyz1 = x[:,t+1], xyz2 = x[:,t]
    const int tid     = threadIdx.x;
    const int wave    = tid >> 5;
    const int lane    = tid & 31;
    const int half    = lane >> 4;         // 0: lanes 0-15, 1: lanes 16-31
    const int l16     = lane & 15;

    const float* xyz1 = x + (((size_t)b * NT + (t + 1)) * NPTS) * 3;
    const float* xyz2 = x + (((size_t)b * NT + t) * NPTS) * 3;
    const int row0 = tileRow * 16;

    // ---------------- Phase A: 16x1024 dot tile via fp32 WMMA ----------------
    // Branchless fragment build: lane loads q[0] and q[1-half] (both always
    // in-bounds), then one v_cndmask supplies the K=3 zero pad. Avoids the
    // exec-mask save/restore branches in the hot loop.
    // A (16x4): lanes 0-15 -> {x1,y1} of row row0+l16 ; lanes 16-31 -> {z1,0}
    v2f afrag;
    {
        const float* p = xyz1 + (size_t)(row0 + l16) * 3 + 2 * half;
        const float pa = p[0];            // half=0: x1 ; half=1: z1
        const float pb = p[1 - half];     // half=0: y1 ; half=1: z1 (unused)
        afrag[0] = pa;
        afrag[1] = half ? 0.0f : pb;
    }
    // each wave covers 16 consecutive 16-wide column tiles
    for (int ct = wave * 16; ct < wave * 16 + 16; ++ct) {
        const int col0 = ct * 16;
        const float* q = xyz2 + (size_t)(col0 + l16) * 3 + 2 * half;
        const float qa = q[0];            // half=0: x2 ; half=1: z2
        const float qb = q[1 - half];     // half=0: y2 ; half=1: z2 (unused)
        v2f bfrag;                        // B (4x16), rows striped across lanes
        bfrag[0] = qa;
        bfrag[1] = half ? 0.0f : qb;
        v8f cfrag = {};
        cfrag = __builtin_amdgcn_wmma_f32_16x16x4_f32(
            false, afrag, false, bfrag, (short)0, cfrag, false, false);
        // D layout: VGPR r: lanes0-15 -> (M=r, N=lane); lanes16-31 -> (M=r+8)
#pragma unroll
        for (int r = 0; r < 8; ++r) {
            const int m = r + 8 * half;
            dotTile[m * NPTS + col0 + l16] = cfrag[r];
        }
    }
    __syncthreads();

    // ---------------- Phase B: top-K + edge MLP + channel max ----------------
    const float bninv = rsqrtf(1.0f + EPSF);
    float acc0 = 0.0f, acc1 = 0.0f;        // channels 2*lane, 2*lane+1

    for (int rr = 0; rr < 4; ++rr) {
        const int m = wave + rr * 4;       // row within tile (0..15)
        const int grow = row0 + m;
        const float c0 = xyz1[grow * 3 + 0];
        const float c1 = xyz1[grow * 3 + 1];
        const float c2 = xyz1[grow * 3 + 2];
        const float n1 = c0 * c0 + c1 * c1 + c2 * c2;

        // per-lane slice of the distance row: cols lane + 32*j
        float dist[32];
#pragma unroll
        for (int j = 0; j < 32; ++j) {
            const int col = lane + 32 * j;
            const float qx = xyz2[col * 3 + 0];
            const float qy = xyz2[col * 3 + 1];
            const float qz = xyz2[col * 3 + 2];
            const float n2 = qx * qx + qy * qy + qz * qz;
            dist[j] = n1 + n2 - 2.0f * dotTile[m * NPTS + col];
        }

        for (int k = 0; k < KNN; ++k) {
            // lane-local argmin over 32 register slots
            float bv = dist[0]; int bj = 0;
#pragma unroll
            for (int j = 1; j < 32; ++j)
                if (dist[j] < bv) { bv = dist[j]; bj = j; }
            // wave32 argmin reduction carrying (value, lane)
            float rv = bv; int rl = lane;
#pragma unroll
            for (int off = 16; off > 0; off >>= 1) {
                const float ov = __shfl_xor(rv, off, 32);
                const int   ol = __shfl_xor(rl, off, 32);
                if (ov < rv || (ov == rv && ol < rl)) { rv = ov; rl = ol; }
            }
            const int wj = __shfl(bj, rl, 32);
            // winner marks its slot (predicated, stays in registers)
#pragma unroll
            for (int j = 0; j < 32; ++j)
                dist[j] = (lane == rl && j == wj) ? 3.4e38f : dist[j];
            const int wincol = rl + 32 * wj;

            // neighbor coords (uniform address -> broadcast load)
            const float nx = xyz2[wincol * 3 + 0];
            const float ny = xyz2[wincol * 3 + 1];
            const float nz = xyz2[wincol * 3 + 2];
            // feat = cat(neigh, center - neigh)
            const float f0 = nx, f1 = ny, f2 = nz;
            const float f3 = c0 - nx, f4 = c1 - ny, f5 = c2 - nz;
#pragma unroll
            for (int cc = 0; cc < 2; ++cc) {
                const int ch = 2 * lane + cc;
                float h = b1[ch];
                h = fmaf(f0, W1[0 * HD + ch], h);
                h = fmaf(f1, W1[1 * HD + ch], h);
                h = fmaf(f2, W1[2 * HD + ch], h);
                h = fmaf(f3, W1[3 * HD + ch], h);
                h = fmaf(f4, W1[4 * HD + ch], h);
                h = fmaf(f5, W1[5 * HD + ch], h);
                h = fmaxf(h, 0.0f);
                h = h * (g1[ch] * bninv) + be1[ch];   // eval-mode BN affine
                if (cc == 0) acc0 = fmaxf(acc0, h);
                else         acc1 = fmaxf(acc1, h);
            }
        }
    }

    // clamp at 0 (reference clamps the pooled max) -> nonneg floats order like
    // their int bit patterns, so signed int atomicMax is a valid float max
    float* ap = acc + ((size_t)t * NB + b) * HD;
    atomicMax((int*)&ap[2 * lane + 0], __float_as_int(fmaxf(acc0, 0.0f)));
    atomicMax((int*)&ap[2 * lane + 1], __float_as_int(fmaxf(acc1, 0.0f)));
}

// ---------------------------------------------------------------------------
// head layer 1: [32,448] @ [448,512] + relu + BN  -> ha [32,512]
// ---------------------------------------------------------------------------
__global__ __launch_bounds__(256) void head1_kernel(
    const float* __restrict__ acc, const float* __restrict__ Wa,
    const float* __restrict__ ba, const float* __restrict__ ga,
    const float* __restrict__ bea, float* __restrict__ ha)
{
    __shared__ float s[(NT - 1) * HD];     // 448
    const int b = blockIdx.x, tid = threadIdx.x;
    for (int d = tid; d < (NT - 1) * HD; d += 256) {
        const int tt = d / HD, ch = d % HD;
        s[d] = acc[((size_t)tt * NB + b) * HD + ch];
    }
    __syncthreads();
    const float bninv = rsqrtf(1.0f + EPSF);
    for (int j = tid; j < 512; j += 256) {
        float v = ba[j];
        for (int d = 0; d < (NT - 1) * HD; ++d)
            v = fmaf(s[d], Wa[(size_t)d * 512 + j], v);
        v = fmaxf(v, 0.0f) * (ga[j] * bninv) + bea[j];
        ha[(size_t)b * 512 + j] = v;
    }
}

// ---------------------------------------------------------------------------
// head layer 2: [32,512] @ [512,256] + relu + BN  -> hb [32,256]
// ---------------------------------------------------------------------------
__global__ __launch_bounds__(256) void head2_kernel(
    const float* __restrict__ ha, const float* __restrict__ Wb,
    const float* __restrict__ bb, const float* __restrict__ gb,
    const float* __restrict__ beb, float* __restrict__ hb)
{
    __shared__ float s[512];
    const int b = blockIdx.x, tid = threadIdx.x;
    for (int d = tid; d < 512; d += 256) s[d] = ha[(size_t)b * 512 + d];
    __syncthreads();
    const float bninv = rsqrtf(1.0f + EPSF);
    for (int j = tid; j < 256; j += 256) {
        float v = bb[j];
        for (int d = 0; d < 512; ++d)
            v = fmaf(s[d], Wb[(size_t)d * 256 + j], v);
        v = fmaxf(v, 0.0f) * (gb[j] * bninv) + beb[j];
        hb[(size_t)b * 256 + j] = v;
    }
}

// ---------------------------------------------------------------------------
// head layer 3: [32,256] @ [256,14] + log_softmax, one wave32 per row
// ---------------------------------------------------------------------------
__global__ __launch_bounds__(32) void head3_kernel(
    const float* __restrict__ hb, const float* __restrict__ Wc,
    const float* __restrict__ bc, float* __restrict__ out)
{
    const int b = blockIdx.x, lane = threadIdx.x;
    float logit = -3.4e38f;
    if (lane < CLS) {
        float v = bc[lane];
        for (int d = 0; d < 256; ++d)
            v = fmaf(hb[(size_t)b * 256 + d], Wc[(size_t)d * CLS + lane], v);
        logit = v;
    }
    float mx = logit;
#pragma unroll
    for (int off = 16; off > 0; off >>= 1)
        mx = fmaxf(mx, __shfl_xor(mx, off, 32));
    float e = (lane < CLS) ? expf(logit - mx) : 0.0f;
#pragma unroll
    for (int off = 16; off > 0; off >>= 1)
        e += __shfl_xor(e, off, 32);
    if (lane < CLS) out[(size_t)b * CLS + lane] = logit - mx - logf(e);
}

// ---------------------------------------------------------------------------
extern "C" void kernel_launch(void* const* d_in, const int* in_sizes, int n_in,
                              void* d_out, int out_size, void* d_ws, size_t ws_size,
                              hipStream_t stream) {
    // setup_inputs() order:
    const float* x   = (const float*)d_in[0];   // [32,8,1024,3]
    // d_in[1] = batch (int64) -- structure is explicit, unused
    const float* W1  = (const float*)d_in[2];
    const float* b1  = (const float*)d_in[3];
    const float* g1  = (const float*)d_in[4];
    const float* be1 = (const float*)d_in[5];
    const float* Wa  = (const float*)d_in[6];
    const float* ba  = (const float*)d_in[7];
    const float* ga  = (const float*)d_in[8];
    const float* bea = (const float*)d_in[9];
    const float* Wb  = (const float*)d_in[10];
    const float* bb  = (const float*)d_in[11];
    const float* gb  = (const float*)d_in[12];
    const float* beb = (const float*)d_in[13];
    const float* Wc  = (const float*)d_in[14];
    const float* bc  = (const float*)d_in[15];
    float* out = (float*)d_out;

    // workspace layout (floats): acc[7*32*64] | ha[32*512] | hb[32*256]
    float* acc = (float*)d_ws;
    float* ha  = acc + (NT - 1) * NB * HD;     // +14336
    float* hb  = ha + NB * 512;                // +16384

    init_acc_kernel<<<((NT - 1) * NB * HD + 255) / 256, 256, 0, stream>>>(acc);

    dim3 egrid(NPTS / 16, NB, NT - 1);         // (64, 32, 7)
    edge_kernel<<<egrid, 128, 0, stream>>>(x, W1, b1, g1, be1, acc);

    head1_kernel<<<NB, 256, 0, stream>>>(acc, Wa, ba, ga, bea, ha);
    head2_kernel<<<NB, 256, 0, stream>>>(ha, Wb, bb, gb, beb, hb);
    head3_kernel<<<NB, 32, 0, stream>>>(hb, Wc, bc, out);
}